// ColorGNNSmall_59287728554038
// MI455X (gfx1250) — compile-verified
//
#include <hip/hip_runtime.h>

#define NEG_SLOPE 0.01f

typedef __attribute__((ext_vector_type(2))) float v2f;
typedef __attribute__((ext_vector_type(8))) float v8f;

// ---------------------------------------------------------------------------
// Graph normalization
// ---------------------------------------------------------------------------
__global__ void k_init_deg(float* __restrict__ deg, int n) {
  int i = blockIdx.x * blockDim.x + threadIdx.x;
  if (i < n) deg[i] = 1.0f;  // self-loop weight
}

__global__ void k_deg_atomic(const long long* __restrict__ col,
                             const float* __restrict__ ew,
                             float* __restrict__ deg, int E) {
  int e = blockIdx.x * blockDim.x + threadIdx.x;
  if (e < E) unsafeAtomicAdd(&deg[(int)col[e]], ew[e]);
}

__global__ void k_dinv(const float* __restrict__ deg, float* __restrict__ dinv, int n) {
  int i = blockIdx.x * blockDim.x + threadIdx.x;
  if (i < n) {
    float d = deg[i];
    dinv[i] = d > 0.0f ? rsqrtf(d) : 0.0f;
  }
}

__global__ void k_norm(const long long* __restrict__ row,
                       const long long* __restrict__ col,
                       const float* __restrict__ ew,
                       const float* __restrict__ dinv,
                       float* __restrict__ nrm, int E) {
  int e = blockIdx.x * blockDim.x + threadIdx.x;
  if (e < E) nrm[e] = dinv[(int)row[e]] * ew[e] * dinv[(int)col[e]];
}

// ---------------------------------------------------------------------------
// GEMM  C[M x NC] = A[M x 128] * B[128 x NC]   via V_WMMA_F32_16X16X4_F32
// One wave per 16x16 tile of C. K = 128, stepped by 4.
// A frag (16x4, f32): lane L holds row m = L&15; VGPR0/1 = K pair;
//   lanes 0-15 -> K = k,k+1 ; lanes 16-31 -> K = k+2,k+3.
// B frag (4x16, f32): lane L holds col n = L&15, same K-half split as A.
// C/D (16x16, f32): VGPR v, lane L -> row = v + 8*(L>>4), col = L&15.
// ---------------------------------------------------------------------------
template <int NC>
__global__ __launch_bounds__(128) void k_gemm_wmma(const float* __restrict__ A,
                                                   const float* __restrict__ B,
                                                   float* __restrict__ C, int M) {
  const int K = 128;
  int wave = (int)((blockIdx.x * blockDim.x + threadIdx.x) >> 5);
  int lane = threadIdx.x & 31;
  const int tilesN = NC / 16;
  int tm = wave / tilesN;
  int tn = wave % tilesN;
  if (tm * 16 >= M) return;  // wave-uniform: whole wave exits before any WMMA

  int m  = lane & 15;
  int kh = (lane >> 4) << 1;  // 0 or 2: K sub-pair for this half-wave

  const float* arow = A + (size_t)(tm * 16 + m) * K;
  const float* bcol = B + tn * 16 + (lane & 15);

  v8f acc = {};
#pragma unroll 4
  for (int k = 0; k < K; k += 4) {
    v2f a;
    a.x = arow[k + kh];
    a.y = arow[k + kh + 1];
    v2f b;
    b.x = bcol[(size_t)(k + kh) * NC];
    b.y = bcol[(size_t)(k + kh + 1) * NC];
    acc = __builtin_amdgcn_wmma_f32_16x16x4_f32(false, a, false, b,
                                                (short)0, acc, false, false);
  }

  int r0 = tm * 16 + ((lane >> 4) << 3);
  int cc = tn * 16 + (lane & 15);
  float* cp = C + (size_t)r0 * NC + cc;
#pragma unroll
  for (int v = 0; v < 8; ++v) cp[(size_t)v * NC] = acc[v];
}

// ---------------------------------------------------------------------------
// Self-loop init: agg[i][f] = xw[i][f] * dinv[i]^2   (norm of self loop)
// ---------------------------------------------------------------------------
__global__ void k_selfloop(const float* __restrict__ xw,
                           const float* __restrict__ dinv,
                           float* __restrict__ agg, int total, int fshift) {
  int i = blockIdx.x * blockDim.x + threadIdx.x;
  if (i < total) {
    float d = dinv[i >> fshift];
    agg[i] = xw[i] * d * d;
  }
}

// ---------------------------------------------------------------------------
// Edge aggregation, F=128: one wave32 per edge, float4 per lane (32*4 = 128)
// ---------------------------------------------------------------------------
__global__ void k_edge_agg128(const long long* __restrict__ row,
                              const long long* __restrict__ col,
                              const float* __restrict__ nrm,
                              const float* __restrict__ xw,
                              float* __restrict__ agg, int E) {
  int e = (int)((blockIdx.x * blockDim.x + threadIdx.x) >> 5);
  int lane = threadIdx.x & 31;
  if (e >= E) return;
  int r = (int)row[e];
  int c = (int)col[e];
  float s = nrm[e];
  const float4* src = reinterpret_cast<const float4*>(xw + (size_t)r * 128) + lane;
  float4 v = *src;
  float* dst = agg + (size_t)c * 128 + lane * 4;
  unsafeAtomicAdd(dst + 0, v.x * s);
  unsafeAtomicAdd(dst + 1, v.y * s);
  unsafeAtomicAdd(dst + 2, v.z * s);
  unsafeAtomicAdd(dst + 3, v.w * s);
}

// ---------------------------------------------------------------------------
// Edge aggregation, F=32: one wave32 per edge, one feature per lane
// ---------------------------------------------------------------------------
__global__ void k_edge_agg32(const long long* __restrict__ row,
                             const long long* __restrict__ col,
                             const float* __restrict__ nrm,
                             const float* __restrict__ xw,
                             float* __restrict__ agg, int E) {
  int e = (int)((blockIdx.x * blockDim.x + threadIdx.x) >> 5);
  int lane = threadIdx.x & 31;
  if (e >= E) return;
  int r = (int)row[e];
  int c = (int)col[e];
  float s = nrm[e];
  float v = xw[(size_t)r * 32 + lane] * s;
  unsafeAtomicAdd(agg + (size_t)c * 32 + lane, v);
}

// ---------------------------------------------------------------------------
// In-place bias + LeakyReLU
// ---------------------------------------------------------------------------
__global__ void k_bias_lrelu(float* __restrict__ h, const float* __restrict__ b,
                             int total, int fmask) {
  int i = blockIdx.x * blockDim.x + threadIdx.x;
  if (i < total) {
    float v = h[i] + b[i & fmask];
    h[i] = v > 0.0f ? v : NEG_SLOPE * v;
  }
}

// ---------------------------------------------------------------------------
// Fused: h2 = leaky_relu(agg2 + b2); out = h2 @ Wc + bc   (32 -> 3)
// ---------------------------------------------------------------------------
__global__ void k_final(const float* __restrict__ agg2, const float* __restrict__ b2,
                        const float* __restrict__ Wc, const float* __restrict__ bc,
                        float* __restrict__ out, int n) {
  int i = blockIdx.x * blockDim.x + threadIdx.x;
  if (i >= n) return;
  const float* a = agg2 + (size_t)i * 32;
  float o0 = bc[0], o1 = bc[1], o2 = bc[2];
#pragma unroll
  for (int f = 0; f < 32; ++f) {
    float v = a[f] + b2[f];
    v = v > 0.0f ? v : NEG_SLOPE * v;
    o0 += v * Wc[f * 3 + 0];
    o1 += v * Wc[f * 3 + 1];
    o2 += v * Wc[f * 3 + 2];
  }
  out[(size_t)i * 3 + 0] = o0;
  out[(size_t)i * 3 + 1] = o1;
  out[(size_t)i * 3 + 2] = o2;
}

// ---------------------------------------------------------------------------
// Host launcher
// ---------------------------------------------------------------------------
extern "C" void kernel_launch(void* const* d_in, const int* in_sizes, int n_in,
                              void* d_out, int out_size, void* d_ws, size_t ws_size,
                              hipStream_t stream) {
  const float*     x   = (const float*)d_in[0];      // [N,128]
  const long long* ei  = (const long long*)d_in[1];  // [2,E] int64
  const float*     ew  = (const float*)d_in[2];      // [E]
  const float*     W1  = (const float*)d_in[3];      // [128,128]
  const float*     b1  = (const float*)d_in[4];      // [128]
  const float*     W2  = (const float*)d_in[5];      // [128,32]
  const float*     b2  = (const float*)d_in[6];      // [32]
  const float*     Wc  = (const float*)d_in[7];      // [32,3]
  const float*     bc  = (const float*)d_in[8];      // [3]
  float*           out = (float*)d_out;              // [N,3]

  const int E = in_sizes[2];
  const int N = in_sizes[0] / 128;

  const long long* row = ei;       // edge_index[0]
  const long long* col = ei + E;   // edge_index[1]

  // workspace layout (floats)
  float* ws   = (float*)d_ws;
  float* deg  = ws;                       // N
  float* dinv = deg + N;                  // N
  float* nrm  = dinv + N;                 // E
  float* xw1  = nrm + E;                  // N*128
  float* agg1 = xw1 + (size_t)N * 128;    // N*128  (becomes h1 in place)
  float* xw2  = xw1;                      // N*32  (reuse: xw1 dead after layer-1 agg)
  float* agg2 = xw1 + (size_t)N * 32;     // N*32  (reuse)
  (void)ws_size; (void)n_in; (void)out_size;

  const int B = 256;
  int nb_nodes  = (N + B - 1) / B;
  int nb_edges  = (E + B - 1) / B;
  int nb_ewave  = ((E * 32) + B - 1) / B;  // wave per edge

  // normalization (shared by both layers)
  k_init_deg<<<nb_nodes, B, 0, stream>>>(deg, N);
  k_deg_atomic<<<nb_edges, B, 0, stream>>>(col, ew, deg, E);
  k_dinv<<<nb_nodes, B, 0, stream>>>(deg, dinv, N);
  k_norm<<<nb_edges, B, 0, stream>>>(row, col, ew, dinv, nrm, E);

  // layer 1: xw1 = x @ W1 ; agg1 = selfloop + scatter ; h1 = lrelu(agg1 + b1)
  {
    int waves = (N / 16) * (128 / 16);
    int blocks = (waves * 32 + 127) / 128;
    k_gemm_wmma<128><<<blocks, 128, 0, stream>>>(x, W1, xw1, N);
  }
  k_selfloop<<<((size_t)N * 128 + B - 1) / B, B, 0, stream>>>(xw1, dinv, agg1, N * 128, 7);
  k_edge_agg128<<<nb_ewave, B, 0, stream>>>(row, col, nrm, xw1, agg1, E);
  k_bias_lrelu<<<((size_t)N * 128 + B - 1) / B, B, 0, stream>>>(agg1, b1, N * 128, 127);

  // layer 2: xw2 = h1 @ W2 ; agg2 = selfloop + scatter
  {
    int waves = (N / 16) * (32 / 16);
    int blocks = (waves * 32 + 127) / 128;
    k_gemm_wmma<32><<<blocks, 128, 0, stream>>>(agg1, W2, xw2, N);
  }
  k_selfloop<<<((size_t)N * 32 + B - 1) / B, B, 0, stream>>>(xw2, dinv, agg2, N * 32, 5);
  k_edge_agg32<<<nb_ewave, B, 0, stream>>>(row, col, nrm, xw2, agg2, E);

  // fused lrelu(·+b2) and 32->3 classifier
  k_final<<<nb_nodes, B, 0, stream>>>(agg2, b2, Wc, bc, out, N);
}